// RBFLayer_27573690040787
// MI455X (gfx1250) — compile-verified
//
#include <hip/hip_runtime.h>
#include <hip/hip_bf16.h>

typedef float v2f __attribute__((ext_vector_type(2)));
typedef float v4f __attribute__((ext_vector_type(4)));
typedef float v8f __attribute__((ext_vector_type(8)));
typedef int   v4i __attribute__((ext_vector_type(4)));

#define D_DIM   256
#define K_DIM   1024
#define BM      64          // rows of x per workgroup (4 row-blocks of 16)
#define RB      4           // row-blocks
#define XP      260         // LDS pitch for x tile (260%64==4 -> conflict-free)
#define CP      260         // LDS pitch for r chunk
#define KCHUNK  256         // K processed per fused chunk

#if defined(__AMDGCN__) && \
    __has_builtin(__builtin_amdgcn_global_load_async_to_lds_b128) && \
    __has_builtin(__builtin_amdgcn_s_wait_asynccnt)
#define USE_ASYNC_LDS 1
typedef __attribute__((address_space(1))) v4i g_v4i;   // global int4
typedef __attribute__((address_space(3))) v4i l_v4i;   // LDS int4
#else
#define USE_ASYNC_LDS 0
#endif

// ---------------------------------------------------------------------------
// Kernel A: per-center data. cnorm[k] = |c_k|^2 ; invdom[k] = 1/dom[k]
// dom[k] = 2 * (sum_{4 nearest j != k} ||c_k - c_j|| * alpha[k]/4)^2
// ---------------------------------------------------------------------------
__global__ __launch_bounds__(256) void rbf_centers_kernel(
    const float* __restrict__ centers, const float* __restrict__ alpha,
    float* __restrict__ cnorm, float* __restrict__ invdom)
{
    __shared__ float ck[D_DIM];
    __shared__ float red[256];
    __shared__ float dist[K_DIM];

    const int k = blockIdx.x;
    const int t = threadIdx.x;

    float cv = centers[(size_t)k * D_DIM + t];
    ck[t]  = cv;
    red[t] = cv * cv;
    __syncthreads();
    for (int s = 128; s > 0; s >>= 1) {
        if (t < s) red[t] += red[t + s];
        __syncthreads();
    }
    if (t == 0) cnorm[k] = red[0];

    for (int i = 0; i < K_DIM / 256; ++i) {
        const int j = i * 256 + t;
        float s = 3.0e38f;                     // self -> +inf (excluded)
        if (j != k) {
            const v4f* cj  = (const v4f*)(centers + (size_t)j * D_DIM);
            const v4f* ckv = (const v4f*)ck;
            s = 0.0f;
            #pragma unroll 8
            for (int d4 = 0; d4 < D_DIM / 4; ++d4) {
                v4f df = ckv[d4] - cj[d4];
                s += df.x * df.x + df.y * df.y + df.z * df.z + df.w * df.w;
            }
            s = fmaxf(s, 0.0f);
        }
        dist[j] = s;
    }
    __syncthreads();

    if (t == 0) {
        float sum = 0.0f;
        for (int p = 0; p < 4; ++p) {          // 4 smallest (P=4)
            float m = 3.0e38f; int mi = 0;
            for (int j = 0; j < K_DIM; ++j) {
                float dv = dist[j];
                if (dv < m) { m = dv; mi = j; }
            }
            sum += sqrtf(m);
            dist[mi] = 3.0e38f;
        }
        float cd  = sum * (alpha[k] * 0.25f);
        float dom = 2.0f * cd * cd;
        invdom[k] = 1.0f / dom;
    }
}

// ---------------------------------------------------------------------------
// Kernel B: fused  r = exp((|x|^2 - 2 xC^T + |c|^2) * invdom) ; out = r @ W
// BM=64 rows per WG; wave w owns column-blocks {2w,2w+1} for all 4 row-blocks
// => each B-fragment feeds 4 WMMAs; centers/W read once per WG.
// WMMA f32 16x16x4 fragment layouts per CDNA5 ISA:
//   A: lane l holds (m = l&15, K = 2*(l>>4)+v) in VGPR v  -> float2
//   B: lane l holds (n = l&15, K = 2*(l>>4)+v) in VGPR v  -> float2
//   C/D: VGPR v, lane l -> (m = v + 8*(l>>4), n = l&15)
// ---------------------------------------------------------------------------
__global__ __launch_bounds__(256) void rbf_main_kernel(
    const float* __restrict__ x, const float* __restrict__ centers,
    const float* __restrict__ W, const float* __restrict__ cnorm,
    const float* __restrict__ invdom, float* __restrict__ out)
{
    __shared__ float xs[BM * XP];
    __shared__ float rs[BM * CP];
    __shared__ float xnorm[BM];

    const int tid   = threadIdx.x;
    const int wave  = tid >> 5;
    const int lane  = tid & 31;
    const int lmod  = lane & 15;
    const int lhalf = lane >> 4;
    const int row0  = blockIdx.x * BM;

    // ---- stage x tile into LDS: 64x256 floats = 4096 float4 ----
#if USE_ASYNC_LDS
    #pragma unroll
    for (int p = 0; p < (BM * D_DIM / 4) / 256; ++p) {
        int li = p * 256 + tid;
        int r  = li >> 6;                  // row (64 float4 per row)
        int c4 = li & 63;
        g_v4i* src = (g_v4i*)(uintptr_t)(x + (size_t)(row0 + r) * D_DIM + c4 * 4);
        l_v4i* dst = (l_v4i*)(unsigned)(uintptr_t)(xs + r * XP + c4 * 4);
        __builtin_amdgcn_global_load_async_to_lds_b128(src, dst, 0, 0);
    }
    __builtin_amdgcn_s_wait_asynccnt(0);
#else
    #pragma unroll
    for (int p = 0; p < (BM * D_DIM / 4) / 256; ++p) {
        int li = p * 256 + tid;
        int r  = li >> 6;
        int c4 = li & 63;
        v4f val = *(const v4f*)(x + (size_t)(row0 + r) * D_DIM + c4 * 4);
        *(v4f*)(xs + r * XP + c4 * 4) = val;
    }
#endif
    __syncthreads();
    if (tid < BM) {
        float s = 0.0f;
        for (int d = 0; d < D_DIM; ++d) { float v = xs[tid * XP + d]; s += v * v; }
        xnorm[tid] = s;
    }
    __syncthreads();

    // persistent output accumulators: [row-block][col-tile 0/1]
    v8f oacc[RB][2];
    #pragma unroll
    for (int rb = 0; rb < RB; ++rb) { oacc[rb][0] = (v8f){}; oacc[rb][1] = (v8f){}; }
    const int on0 = wave * 32 + lmod;       // out column, tile 0
    const int on1 = on0 + 16;               // out column, tile 1

    for (int kc = 0; kc < K_DIM; kc += KCHUNK) {
        // ======== GEMM1: dot tiles for centers [kc+32w, kc+32w+32) ========
        v8f acc[RB][2];
        #pragma unroll
        for (int rb = 0; rb < RB; ++rb) { acc[rb][0] = (v8f){}; acc[rb][1] = (v8f){}; }

        const int n0 = kc + wave * 32 + lmod;
        const int n1 = n0 + 16;
        const float* c0p = centers + (size_t)n0 * D_DIM + 2 * lhalf;
        const float* c1p = centers + (size_t)n1 * D_DIM + 2 * lhalf;
        const float* xap = xs + lmod * XP + 2 * lhalf;

        #pragma unroll 2
        for (int d = 0; d < D_DIM; d += 4) {
            v2f b0 = *(const v2f*)(c0p + d);
            v2f b1 = *(const v2f*)(c1p + d);
            #pragma unroll
            for (int rb = 0; rb < RB; ++rb) {
                v2f a = *(const v2f*)(xap + rb * (16 * XP) + d);
                acc[rb][0] = __builtin_amdgcn_wmma_f32_16x16x4_f32(
                                 false, a, false, b0, (short)0, acc[rb][0], false, false);
                acc[rb][1] = __builtin_amdgcn_wmma_f32_16x16x4_f32(
                                 false, a, false, b1, (short)0, acc[rb][1], false, false);
            }
        }

        // ======== epilogue: r = exp(max(xn + cn - 2*dot, 0) * invdom) ======
        const float cn0 = cnorm[n0],  cn1 = cnorm[n1];
        const float id0 = invdom[n0], id1 = invdom[n1];
        const int   ncol = wave * 32 + lmod;      // local column in chunk
        #pragma unroll
        for (int rb = 0; rb < RB; ++rb) {
            #pragma unroll
            for (int v = 0; v < 8; ++v) {
                int   m   = rb * 16 + v + 8 * lhalf;
                float xnv = xnorm[m];
                float s0  = fmaxf(xnv + cn0 - 2.0f * acc[rb][0][v], 0.0f);
                float s1  = fmaxf(xnv + cn1 - 2.0f * acc[rb][1][v], 0.0f);
                rs[m * CP + ncol]      = __expf(s0 * id0);
                rs[m * CP + ncol + 16] = __expf(s1 * id1);
            }
        }
        __syncthreads();

        // ======== GEMM2 partial: oacc += r_chunk @ W[kc:kc+256, :] ========
        const float* rap = rs + lmod * CP + 2 * lhalf;
        const float* w0p = W + (size_t)(kc + 2 * lhalf) * D_DIM + on0;
        const float* w1p = w0p + 16;

        #pragma unroll 2
        for (int d = 0; d < KCHUNK; d += 4) {
            v2f b0, b1;
            b0.x = w0p[(size_t)d * D_DIM];
            b0.y = w0p[(size_t)d * D_DIM + D_DIM];
            b1.x = w1p[(size_t)d * D_DIM];
            b1.y = w1p[(size_t)d * D_DIM + D_DIM];
            #pragma unroll
            for (int rb = 0; rb < RB; ++rb) {
                v2f a = *(const v2f*)(rap + rb * (16 * CP) + d);
                oacc[rb][0] = __builtin_amdgcn_wmma_f32_16x16x4_f32(
                                  false, a, false, b0, (short)0, oacc[rb][0], false, false);
                oacc[rb][1] = __builtin_amdgcn_wmma_f32_16x16x4_f32(
                                  false, a, false, b1, (short)0, oacc[rb][1], false, false);
            }
        }
        __syncthreads();   // rs reused next chunk
    }

    // ---- store output tiles ----
    #pragma unroll
    for (int rb = 0; rb < RB; ++rb) {
        #pragma unroll
        for (int v = 0; v < 8; ++v) {
            int m = rb * 16 + v + 8 * lhalf;
            out[(size_t)(row0 + m) * D_DIM + on0] = oacc[rb][0][v];
            out[(size_t)(row0 + m) * D_DIM + on1] = oacc[rb][1][v];
        }
    }
}

// ---------------------------------------------------------------------------
extern "C" void kernel_launch(void* const* d_in, const int* in_sizes, int n_in,
                              void* d_out, int out_size, void* d_ws, size_t ws_size,
                              hipStream_t stream)
{
    const float* x       = (const float*)d_in[0];   // [B, 256]
    const float* centers = (const float*)d_in[1];   // [1024, 256]
    const float* alpha   = (const float*)d_in[2];   // [1024]
    const float* W       = (const float*)d_in[3];   // [1024, 256]
    float*       out     = (float*)d_out;           // [B, 256]

    float* cnorm  = (float*)d_ws;                   // [1024]
    float* invdom = cnorm + K_DIM;                  // [1024]

    const int Bn = in_sizes[0] / D_DIM;             // 16384

    rbf_centers_kernel<<<K_DIM, 256, 0, stream>>>(centers, alpha, cnorm, invdom);
    rbf_main_kernel<<<Bn / BM, 256, 0, stream>>>(x, centers, W, cnorm, invdom, out);
}